// MambaLikeBlock_33483565039899
// MI455X (gfx1250) — compile-verified
//
#include <hip/hip_runtime.h>
#include <math.h>

// ---------------------------------------------------------------------------
// Problem constants (match reference)
// ---------------------------------------------------------------------------
#define D_MODEL 256
#define D_INNER 512
#define G3      768          // 3 * D_MODEL
#define BATCH   8
#define SEQ_T   4096
#define MTOT    (BATCH * SEQ_T)   // 32768 rows

#define NWG_SCAN 16          // workgroups in the scan kernel
#define JW       16          // hidden columns owned per scan workgroup

typedef float v2f __attribute__((ext_vector_type(2)));
typedef float v8f __attribute__((ext_vector_type(8)));
typedef int   v4i __attribute__((ext_vector_type(4)));
typedef unsigned int u32x4 __attribute__((ext_vector_type(4)));
typedef unsigned int u32x8 __attribute__((ext_vector_type(8)));

#if defined(__gfx1250__) && __has_builtin(__builtin_amdgcn_global_load_async_to_lds_b128)
#define HAVE_ASYNC_LDS 1
typedef __attribute__((address_space(1))) v4i* gv4i_ptr;  // global int4*
typedef __attribute__((address_space(3))) v4i* lv4i_ptr;  // LDS int4*
#endif

// D = A(16x4 f32) * B(4x16 f32) + C(16x16 f32), wave32 WMMA
static __device__ __forceinline__ v8f wmma_f32(v2f a, v2f b, v8f c) {
  return __builtin_amdgcn_wmma_f32_16x16x4_f32(
      /*neg_a=*/false, a, /*neg_b=*/false, b,
      /*c_mod=*/(short)0, c, /*reuse_a=*/false, /*reuse_b=*/false);
}

static __device__ __forceinline__ float gelu_exact(float v) {
  return 0.5f * v * (1.0f + erff(v * 0.70710678118654752f));
}

static __device__ __forceinline__ void wait_asynccnt0() {
#if defined(__gfx1250__)
#if __has_builtin(__builtin_amdgcn_s_wait_asynccnt)
  __builtin_amdgcn_s_wait_asynccnt(0);
#else
  asm volatile("s_wait_asynccnt 0x0" ::: "memory");
#endif
#endif
}

static __device__ __forceinline__ void wait_tensorcnt0() {
#if defined(__gfx1250__)
#if __has_builtin(__builtin_amdgcn_s_wait_tensorcnt)
  __builtin_amdgcn_s_wait_tensorcnt(0);
#else
  asm volatile("s_wait_tensorcnt 0x0" ::: "memory");
#endif
#endif
}

// ---------------------------------------------------------------------------
// Kernel 0: zero the recurrent state + sync counter (every launch, so the
// graph-replayed call sequence is deterministic).
// ---------------------------------------------------------------------------
__global__ void init_state_kernel(float* hstate, unsigned int* counter) {
  int i = blockIdx.x * blockDim.x + threadIdx.x;
  if (i < 2 * BATCH * D_MODEL) hstate[i] = 0.0f;
  if (i == 0) *counter = 0u;
}

// ---------------------------------------------------------------------------
// Kernel 1: fused  h = gelu(X @ W1^T + b1)  (kept in LDS)
//                 gx = h @ W_ih^T + b_ih    (written to workspace)
// One block = 256 threads = 8 waves, handles 16 rows of M.
// A operand fetched as a per-lane b64 at (k0 + 2*hi): the f32-WMMA A layout
// needs exactly {K0,K1} (lanes 0-15) / {K2,K3} (lanes 16-31), contiguous.
// ---------------------------------------------------------------------------
__global__ __launch_bounds__(256) void fused_gemm_kernel(
    const float* __restrict__ X, const float* __restrict__ W1,
    const float* __restrict__ b1, const float* __restrict__ W_ih,
    const float* __restrict__ b_ih, float* __restrict__ gx) {
  __shared__ __align__(16) float hs[16][D_INNER];  // 32 KB

  const int lane  = threadIdx.x & 31;
  const int wave  = threadIdx.x >> 5;
  const int row16 = lane & 15;       // M row (A/C) or N col (B/C)
  const int hi    = lane >> 4;       // upper half-wave flag
  const int m0    = blockIdx.x * 16; // this block's M base

  // ---------------- Phase A: X[16,256] @ W1^T -> hs[16,512] ----------------
  {
    const int nb = wave * 64;
    v8f acc[4] = {};
    const float* Xrow = X + (size_t)(m0 + row16) * D_MODEL + 2 * hi;
#pragma unroll 4
    for (int k0 = 0; k0 < D_MODEL; k0 += 4) {
      __builtin_prefetch(Xrow + k0 + 64, 0, 3);
      v2f a = *(const v2f*)(Xrow + k0);   // single b64, no selects
#pragma unroll
      for (int nn = 0; nn < 4; ++nn) {
        const float* Wrow = W1 + (size_t)(nb + nn * 16 + row16) * D_MODEL + k0;
        float4 b4 = *(const float4*)Wrow;
        v2f b;  // B VGPR0: rows K0 (lanes 0-15) / K1 (lanes 16-31); VGPR1: K2/K3
        b.x = hi ? b4.y : b4.x;
        b.y = hi ? b4.w : b4.z;
        acc[nn] = wmma_f32(a, b, acc[nn]);
      }
    }
    // epilogue: + b1, exact GELU, park in LDS
#pragma unroll
    for (int nn = 0; nn < 4; ++nn) {
      const int n = nb + nn * 16 + row16;
      const float bias = b1[n];
#pragma unroll
      for (int r = 0; r < 8; ++r) {
        const int m = r + hi * 8;  // C layout: vgpr r -> M=r (lo) / M=8+r (hi)
        hs[m][n] = gelu_exact(acc[nn][r] + bias);
      }
    }
  }
  __syncthreads();

  // ---------------- Phase B: hs[16,512] @ W_ih^T -> gx[16,768] -------------
  {
    const int nb = wave * 96;
    v8f acc[6] = {};
#pragma unroll 2
    for (int k0 = 0; k0 < D_INNER; k0 += 4) {
      v2f a = *(const v2f*)(&hs[row16][k0 + 2 * hi]);  // ds_load_b64, no selects
#pragma unroll
      for (int nn = 0; nn < 6; ++nn) {
        const float* Wrow = W_ih + (size_t)(nb + nn * 16 + row16) * D_INNER + k0;
        float4 b4 = *(const float4*)Wrow;
        v2f b;
        b.x = hi ? b4.y : b4.x;
        b.y = hi ? b4.w : b4.z;
        acc[nn] = wmma_f32(a, b, acc[nn]);
      }
    }
#pragma unroll
    for (int nn = 0; nn < 6; ++nn) {
      const int n = nb + nn * 16 + row16;
      const float bias = b_ih[n];
#pragma unroll
      for (int r = 0; r < 8; ++r) {
        const int m = m0 + r + hi * 8;
        gx[(size_t)m * G3 + n] = acc[nn][r] + bias;
      }
    }
  }
}

// ---------------------------------------------------------------------------
// Kernel 2: sequential GRU scan. 16 persistent workgroups x 128 threads.
// WG owns hidden columns [jb, jb+16); LDS holds the 48 matching W_hh rows
// (r/z/n for the same j -> gate combine is WG-local). W_hh slice is staged
// by the Tensor Data Mover (3 D# descriptors, one per gate); the per-step
// h_prev broadcast uses async global->LDS loads (ASYNCcnt).
// ---------------------------------------------------------------------------
__global__ __launch_bounds__(128) void gru_scan_kernel(
    const float* __restrict__ gx, const float* __restrict__ W_hh,
    const float* __restrict__ b_hh, float* __restrict__ hstate,  // [2][8][256]
    unsigned int* __restrict__ counter, float* __restrict__ out) {
  __shared__ __align__(16) float Wl[3][JW][D_MODEL];  // 48 KB, resident all steps
  __shared__ __align__(16) float hp[BATCH][D_MODEL];  // 8 KB, h_prev broadcast
  __shared__ float gh[3][BATCH][JW];                  // 1.5 KB gate pre-acts

  const int tid   = threadIdx.x;
  const int lane  = tid & 31;
  const int wave  = tid >> 5;
  const int row16 = lane & 15;
  const int hi    = lane >> 4;
  const int jb    = blockIdx.x * JW;

  // ---- Preload this WG's W_hh slice (rows {g*256 + jb + jj}) via TDM ----
#if defined(__gfx1250__)
  if (wave == 0) {
    // Issue one tensor_load_to_lds per gate: 2D tile 256x16 f32, row stride
    // 256 elements, contiguous 16 KB landing zone in LDS.
#pragma unroll
    for (int g = 0; g < 3; ++g) {
      const unsigned long long gaddr =
          (unsigned long long)(W_hh + (size_t)(g * D_MODEL + jb) * D_MODEL);
      const unsigned int laddr =
          (unsigned int)(unsigned long long)&Wl[g][0][0];  // flat->LDS offset
      u32x4 g0;
      g0.x = 1u;                                   // count=1, user descriptor
      g0.y = laddr;                                // lds_addr
      g0.z = (unsigned int)gaddr;                  // global_addr[31:0]
      g0.w = (unsigned int)(gaddr >> 32) | (2u << 30);  // addr[56:32] | type=2
      u32x8 g1;
      g1[0] = 2u << 16;          // workgroup_mask=0 | data_size=4B
      g1[1] = (unsigned int)D_MODEL << 16;  // tensor_dim0[15:0]=256
      g1[2] = (unsigned int)JW << 16;       // tensor_dim0[31:16]=0 | tensor_dim1=16
      g1[3] = (unsigned int)D_MODEL << 16;  // tensor_dim1[31:16]=0 | tile_dim0=256
      g1[4] = (unsigned int)JW;             // tile_dim1=16 | tile_dim2=0
      g1[5] = (unsigned int)D_MODEL;        // tensor_dim0_stride[31:0]=256
      g1[6] = 0u;                           // stride0[47:32] | stride1[15:0]
      g1[7] = 0u;                           // tensor_dim1_stride[47:16]
      asm volatile("tensor_load_to_lds %0, %1" ::"s"(g0), "s"(g1) : "memory");
    }
    wait_tensorcnt0();
  }
#else
  for (int i = tid; i < 3 * JW * D_MODEL; i += 128) {
    const int g  = i / (JW * D_MODEL);
    const int r  = i - g * (JW * D_MODEL);
    const int jj = r / D_MODEL;
    const int k  = r - jj * D_MODEL;
    Wl[g][jj][k] = W_hh[(size_t)(g * D_MODEL + jb + jj) * D_MODEL + k];
  }
#endif
  __syncthreads();

  float* hpf = &hp[0][0];

  for (int t = 0; t < SEQ_T; ++t) {
    const int cur = t & 1, nxt = cur ^ 1;

    // Broadcast h_prev from global scratch into LDS.
    const float* hsrc = hstate + cur * (BATCH * D_MODEL);
#if defined(HAVE_ASYNC_LDS)
    for (int i = tid * 4; i < BATCH * D_MODEL; i += 128 * 4) {
      __builtin_amdgcn_global_load_async_to_lds_b128(
          (gv4i_ptr)(unsigned long long)(hsrc + i),
          (lv4i_ptr)(unsigned int)(unsigned long long)(hpf + i), 0, 0);
    }
    wait_asynccnt0();
#else
    for (int i = tid; i < BATCH * D_MODEL; i += 128) hpf[i] = hsrc[i];
#endif
    __syncthreads();

    // Waves 0..2: gh[g] = h_prev @ W_hh[g-slice]^T via f32 WMMA (K=256).
    if (wave < 3) {
      const int g = wave;
      v8f acc = {};
#pragma unroll 8
      for (int k0 = 0; k0 < D_MODEL; k0 += 4) {
        v2f a;
        if (row16 < BATCH) {
          a = *(const v2f*)(&hp[row16][k0 + 2 * hi]);
        } else {
          a.x = 0.0f; a.y = 0.0f;   // pad rows 8..15 of the M=16 tile
        }
        float4 b4 = *(const float4*)(&Wl[g][row16][k0]);
        v2f b;
        b.x = hi ? b4.y : b4.x;
        b.y = hi ? b4.w : b4.z;
        acc = wmma_f32(a, b, acc);
      }
      if (hi == 0) {  // lanes 0..15 hold M = 0..7 in VGPRs 0..7
#pragma unroll
        for (int r = 0; r < BATCH; ++r) gh[g][r][row16] = acc[r];
      }
    }
    __syncthreads();

    // Gate combine: 128 threads <-> 8 batches x 16 columns.
    {
      const int b  = tid >> 4;
      const int j  = tid & 15;
      const int jg = jb + j;
      const size_t gxb = ((size_t)b * SEQ_T + t) * G3;
      const float hprev = hp[b][jg];
      const float hr = gh[0][b][j] + b_hh[jg];
      const float hz = gh[1][b][j] + b_hh[D_MODEL + jg];
      const float hn = gh[2][b][j] + b_hh[2 * D_MODEL + jg];
      const float xr = gx[gxb + jg];
      const float xz = gx[gxb + D_MODEL + jg];
      const float xn = gx[gxb + 2 * D_MODEL + jg];
      const float rg = 1.0f / (1.0f + __expf(-(xr + hr)));
      const float zg = 1.0f / (1.0f + __expf(-(xz + hz)));
      const float ng = tanhf(xn + rg * hn);
      const float hnew = (1.0f - zg) * ng + zg * hprev;
      hstate[nxt * (BATCH * D_MODEL) + b * D_MODEL + jg] = hnew;
      out[((size_t)b * SEQ_T + t) * D_MODEL + jg] = hnew;
    }

    // Grid-wide step barrier: release writes, arrive, spin, acquire.
    __builtin_amdgcn_fence(__ATOMIC_RELEASE, "agent");
    __syncthreads();
    if (tid == 0) {
      __hip_atomic_fetch_add(counter, 1u, __ATOMIC_RELAXED,
                             __HIP_MEMORY_SCOPE_AGENT);
      const unsigned int target = (unsigned int)(t + 1) * NWG_SCAN;
      while (__hip_atomic_load(counter, __ATOMIC_RELAXED,
                               __HIP_MEMORY_SCOPE_AGENT) < target) {
        __builtin_amdgcn_s_sleep(1);
      }
    }
    __syncthreads();
    __builtin_amdgcn_fence(__ATOMIC_ACQUIRE, "agent");
  }
}

// ---------------------------------------------------------------------------
// Host-side launch. Workspace layout:
//   [0, 96MB)           gx : float[32768][768]
//   [96MB, +16KB)       hstate : float[2][8][256]
//   then                counter : uint32
// ---------------------------------------------------------------------------
extern "C" void kernel_launch(void* const* d_in, const int* in_sizes, int n_in,
                              void* d_out, int out_size, void* d_ws,
                              size_t ws_size, hipStream_t stream) {
  (void)in_sizes; (void)n_in; (void)out_size; (void)ws_size;
  const float* x    = (const float*)d_in[0];
  const float* W1   = (const float*)d_in[1];
  const float* b1   = (const float*)d_in[2];
  const float* W_ih = (const float*)d_in[3];
  const float* b_ih = (const float*)d_in[4];
  const float* W_hh = (const float*)d_in[5];
  const float* b_hh = (const float*)d_in[6];
  float* out = (float*)d_out;

  char* ws = (char*)d_ws;
  const size_t gx_bytes = (size_t)MTOT * G3 * sizeof(float);  // 100663296
  float* gx = (float*)ws;
  float* hstate = (float*)(ws + gx_bytes);
  unsigned int* counter =
      (unsigned int*)(ws + gx_bytes + 2 * BATCH * D_MODEL * sizeof(float));

  init_state_kernel<<<(2 * BATCH * D_MODEL + 255) / 256, 256, 0, stream>>>(
      hstate, counter);
  fused_gemm_kernel<<<MTOT / 16, 256, 0, stream>>>(x, W1, b1, W_ih, b_ih, gx);
  gru_scan_kernel<<<NWG_SCAN, 128, 0, stream>>>(gx, W_hh, b_hh, hstate,
                                                counter, out);
}